// CustomCrossEntropyLoss_47450798686792
// MI455X (gfx1250) — compile-verified
//
#include <hip/hip_runtime.h>
#include <stdint.h>

// -------- problem constants --------
constexpr int kN        = 4194304;
constexpr int kC        = 10;
constexpr int BLOCK     = 256;          // 8 waves (wave32)
constexpr int TILE_ROWS = 512;          // 512 rows * 40B = 20480 B per tile
constexpr int TILE_FLT  = TILE_ROWS * kC;            // 5120 floats
constexpr int CHUNKS    = (TILE_FLT * 4) / (BLOCK * 16);  // 5 x b128 per thread
constexpr int NTILES    = kN / TILE_ROWS;            // 8192 (exact)
constexpr int NBLOCKS   = 1024;                      // 8 tiles per block

typedef int v4i __attribute__((ext_vector_type(4)));

// -------- CDNA5 async global->LDS copy (16B per lane) --------
__device__ __forceinline__ void async_copy_b128(const float* gsrc, float* ldst) {
#if defined(__gfx1250__) && __has_builtin(__builtin_amdgcn_global_load_async_to_lds_b128)
  __builtin_amdgcn_global_load_async_to_lds_b128(
      (__attribute__((address_space(1))) v4i*)(uintptr_t)gsrc,
      (__attribute__((address_space(3))) v4i*)(uint32_t)(uintptr_t)ldst,
      /*imm offset*/ 0, /*cpol*/ 0);
#else
  uint64_t ga = (uint64_t)(uintptr_t)gsrc;
  uint32_t la = (uint32_t)(uintptr_t)ldst;   // generic LDS addr low 32 bits == DS offset
  asm volatile("global_load_async_to_lds_b128 %0, %1, off"
               :: "v"(la), "v"(ga) : "memory");
#endif
}

#if defined(__gfx1250__) && __has_builtin(__builtin_amdgcn_s_wait_asynccnt)
#define WAIT_ASYNC(n) __builtin_amdgcn_s_wait_asynccnt(n)
#else
#define WAIT_ASYNC(n) asm volatile("s_wait_asynccnt %0" :: "n"(n) : "memory")
#endif

// -------- pass 1: per-block partial sums of diff^2 and NLL --------
__global__ void __launch_bounds__(BLOCK)
ce_partials(const float* __restrict__ logits,
            const int*   __restrict__ labels,
            float*       __restrict__ partials) {
  __shared__ float buf[2][TILE_FLT];       // 2 x 20 KB double buffer
  const int tid    = threadIdx.x;
  const int stride = gridDim.x;

  int   isq = 0;      // exact integer sum of (argmax - label)^2
  float nll = 0.0f;

  auto issue_copy = [&](int tile, int b) {
    const float* g = logits + (size_t)tile * TILE_FLT;
#pragma unroll
    for (int j = 0; j < CHUNKS; ++j) {
      const int off = (tid + BLOCK * j) * 4;   // float index of this 16B chunk
      async_copy_b128(g + off, &buf[b][off]);
    }
  };

  int t = blockIdx.x;
  issue_copy(t, 0);                         // prologue (NBLOCKS <= NTILES always)
  int parity = 0;
  for (; t < NTILES; t += stride, parity ^= 1) {
    const int tn = t + stride;
    if (tn < NTILES) {                      // overlap next tile's copy with compute
      issue_copy(tn, parity ^ 1);
      WAIT_ASYNC(CHUNKS);                   // current tile's 5 copies done; next 5 in flight
    } else {
      WAIT_ASYNC(0);
    }
    __syncthreads();                        // all waves' chunks of this tile visible

    // this thread's 2 rows (80B @ 16B alignment)
    const float* x   = &buf[parity][tid * 2 * kC];
    const int rbase  = t * TILE_ROWS + 2 * tid;
    const int2 lab   = *(const int2*)(labels + rbase);
#pragma unroll
    for (int r = 0; r < 2; ++r) {
      const float* v = x + r * kC;
      const int lbl  = (r == 0) ? lab.x : lab.y;
      float m = v[0]; int am = 0;
#pragma unroll
      for (int c = 1; c < kC; ++c)          // strict '>' => first-max, matches jnp.argmax
        if (v[c] > m) { m = v[c]; am = c; }
      float s = 0.0f;
#pragma unroll
      for (int c = 0; c < kC; ++c)
        s += __expf(v[c] - m);
      const float lse = m + __logf(s);      // logsumexp
      nll += lse - v[lbl];
      const int d = am - lbl;
      isq += d * d;
    }
    __syncthreads();                        // buf[parity] free before tile t+2 overwrites it
  }

  // deterministic in-block tree reduction
  __shared__ int   sred[BLOCK];
  __shared__ float fred[BLOCK];
  sred[tid] = isq; fred[tid] = nll;
  __syncthreads();
  for (int w = BLOCK / 2; w > 0; w >>= 1) {
    if (tid < w) { sred[tid] += sred[tid + w]; fred[tid] += fred[tid + w]; }
    __syncthreads();
  }
  if (tid == 0) {
    partials[2 * blockIdx.x]     = (float)sred[0];   // <= ~331k, exact in fp32
    partials[2 * blockIdx.x + 1] = fred[0];
  }
}

// -------- pass 2: deterministic fixed-order finalize in double --------
__global__ void __launch_bounds__(BLOCK)
ce_finalize(const float* __restrict__ partials, float* __restrict__ out) {
  __shared__ double dsq[BLOCK], dnl[BLOCK];
  double s = 0.0, n = 0.0;
  for (int i = threadIdx.x; i < NBLOCKS; i += BLOCK) {
    s += (double)partials[2 * i];
    n += (double)partials[2 * i + 1];
  }
  dsq[threadIdx.x] = s; dnl[threadIdx.x] = n;
  __syncthreads();
  for (int w = BLOCK / 2; w > 0; w >>= 1) {
    if (threadIdx.x < w) {
      dsq[threadIdx.x] += dsq[threadIdx.x + w];
      dnl[threadIdx.x] += dnl[threadIdx.x + w];
    }
    __syncthreads();
  }
  if (threadIdx.x == 0) {
    const double invN = 1.0 / (double)kN;
    const double rms  = sqrt(dsq[0] * invN);
    const double ce   = dnl[0] * invN;
    out[0] = (float)(rms * ce);
  }
}

extern "C" void kernel_launch(void* const* d_in, const int* in_sizes, int n_in,
                              void* d_out, int out_size, void* d_ws, size_t ws_size,
                              hipStream_t stream) {
  const float* logits   = (const float*)d_in[0];
  const int*   labels   = (const int*)d_in[1];
  float*       out      = (float*)d_out;
  float*       partials = (float*)d_ws;    // 2 * NBLOCKS floats = 8 KB

  ce_partials<<<NBLOCKS, BLOCK, 0, stream>>>(logits, labels, partials);
  ce_finalize<<<1, BLOCK, 0, stream>>>(partials, out);
}